// RetrievalModule_15573551415524
// MI455X (gfx1250) — compile-verified
//
#include <hip/hip_runtime.h>
#include <hip/hip_bf16.h>
#include <math.h>

typedef float v2f __attribute__((ext_vector_type(2)));
typedef float v8f __attribute__((ext_vector_type(8)));

// D = A(16x4 f32) * B(4x16 f32) + C(16x16 f32)
#define WMMA4(a, b, c) \
  __builtin_amdgcn_wmma_f32_16x16x4_f32(false, (a), false, (b), (short)0, (c), false, false)

namespace {
constexpr int Bq = 2048;
constexpr int Nn = 65536;
constexpr int PD = 512;
constexpr int TD = 256;
constexpr int Hh = 64;
constexpr int RD = 128;
constexpr int NCHUNK = 32;           // sim column chunks
constexpr int CHUNK = Nn / NCHUNK;   // 2048 columns per chunk (8 waves x 256)
}

// ---------------------------------------------------------------------------
// Shared projection GEMM: out = act(concat(X1,X2) @ W^T + bias)
// MODE 0: bias + L2 normalize rows (OF must be 128); writes row-major and/or
//         paired-transposed ( element (row n, col c) -> outPT[((c>>1)*rows+n)*2+(c&1)] )
// MODE 1: bias + ReLU, row-major; optional has-valid fallback (z_tilde select)
// block: 256 threads (8 waves), 16 rows per block; WMMA f32 16x16x4
// ---------------------------------------------------------------------------
template <int MODE>
__global__ __launch_bounds__(256) void fused_linear_kernel(
    const float* __restrict__ X1, int w1, const float* __restrict__ X2, int w2,
    const float* __restrict__ W, const float* __restrict__ bias, int OF,
    int rowsTotal, float* __restrict__ outRM, float* __restrict__ outPT,
    const float* __restrict__ hv, const float* __restrict__ fallback) {
  extern __shared__ float smem[];
  const int IN = w1 + w2;
  float* sA = smem;                  // [16][IN]
  float* sOut = smem + 16 * IN;      // [16][128]  (MODE 0 only)
  float* sScale = sOut + 16 * 128;   // [16]

  const int tid = threadIdx.x;
  const int wave = tid >> 5;
  const int lane = tid & 31;
  const int rowbase = blockIdx.x * 16;

  // stage the 16-row A tile in LDS (coalesced)
  for (int idx = tid; idx < 16 * IN; idx += 256) {
    int r = idx / IN, c = idx - r * IN;
    float v = (c < w1) ? X1[(size_t)(rowbase + r) * w1 + c]
                       : X2[(size_t)(rowbase + r) * w2 + (c - w1)];
    sA[idx] = v;
  }
  __syncthreads();

  const int m = lane & 15;                 // row (A) / col (B,C) within tile
  const int kline = (lane >> 4) << 1;      // K sub-line: 0 or 2
  const int halfsel = (lane >> 4) << 3;    // C rows: 0..7 or 8..15

  const int ntiles = OF >> 4;
  for (int ct = wave; ct < ntiles; ct += 8) {
    const int cbase = ct << 4;
    const float* wrow = W + (size_t)(cbase + m) * IN;  // B[kk][n] = W[c][kk]
    const float* arow = sA + m * IN;
    v8f acc = {};
#pragma unroll 4
    for (int kk = 0; kk < IN; kk += 4) {
      v2f a = *(const v2f*)(arow + kk + kline);
      v2f b = *(const v2f*)(wrow + kk + kline);
      acc = WMMA4(a, b, acc);
    }
    float bv = bias[cbase + m];
    if (MODE == 0) {
#pragma unroll
      for (int r = 0; r < 8; ++r) {
        int mm = r + halfsel;
        sOut[mm * 128 + cbase + m] = acc[r] + bv;
      }
    } else {
#pragma unroll
      for (int r = 0; r < 8; ++r) {
        size_t grow = (size_t)(rowbase + r + halfsel);
        float v = fmaxf(acc[r] + bv, 0.0f);
        if (hv && hv[grow] == 0.0f) v = fallback[grow * (size_t)OF + cbase + m];
        outRM[grow * (size_t)OF + cbase + m] = v;
      }
    }
  }

  if (MODE == 0) {
    __syncthreads();
    if (tid < 16) {
      float s = 0.f;
      for (int c = 0; c < 128; ++c) {
        float v = sOut[tid * 128 + c];
        s += v * v;
      }
      sScale[tid] = 1.0f / fmaxf(sqrtf(s), 1e-12f);
    }
    __syncthreads();
    for (int idx = tid; idx < 16 * 128; idx += 256) {
      int r = idx >> 7, c = idx & 127;
      float v = sOut[idx] * sScale[r];
      if (outRM) outRM[(size_t)(rowbase + r) * 128 + c] = v;
      if (outPT)
        outPT[(((size_t)(c >> 1)) * (size_t)rowsTotal + (rowbase + r)) * 2 + (c & 1)] = v;
    }
  }
}

// ---------------------------------------------------------------------------
// sim = q @ k^T fused with per-chunk top-16 (never materializes [B,N]).
// grid (B/16, NCHUNK); block 256 = 8 waves; wave covers 16 rows x 256 cols.
// ---------------------------------------------------------------------------
__global__ __launch_bounds__(256) void sim_topk_kernel(
    const float* __restrict__ qn, const float* __restrict__ kP,
    const unsigned char* __restrict__ mask, float* __restrict__ candV,
    int* __restrict__ candI) {
  __shared__ float sQ[16 * 128];
  __shared__ float sTile[8][16 * 17];
  __shared__ float sTV[8][16][16];
  __shared__ int sTI[8][16][16];
  __shared__ float sMV[16][16];
  __shared__ int sMI[16][16];

  const int tid = threadIdx.x;
  const int wave = tid >> 5;
  const int lane = tid & 31;
  const int rowbase = blockIdx.x * 16;
  const int wcolbase = blockIdx.y * CHUNK + wave * 256;

  for (int idx = tid; idx < 16 * 128; idx += 256)
    sQ[idx] = qn[(size_t)(rowbase + (idx >> 7)) * 128 + (idx & 127)];
  for (int idx = tid; idx < 8 * 16 * 16; idx += 256) {
    (&sTV[0][0][0])[idx] = -3.0e38f;
    (&sTI[0][0][0])[idx] = 0;
  }
  __syncthreads();

  const int m = lane & 15;
  const int kline = (lane >> 4) << 1;
  const int khalf = lane >> 4;
  const int halfsel = (lane >> 4) << 3;

  // Preload all 32 A fragments (16 rows x 128 features) into registers
  v2f aq[32];
#pragma unroll
  for (int j = 0; j < 32; ++j)
    aq[j] = *(const v2f*)(sQ + m * 128 + 4 * j + kline);

  const size_t stride2N = 2 * (size_t)Nn;

  for (int ct = 0; ct < 16; ++ct) {
    const int ncol = wcolbase + ct * 16;
    const float* kptr = kP + 2 * (size_t)(ncol + m);  // paired-transposed k
    v8f acc = {};
#pragma unroll
    for (int j = 0; j < 32; ++j) {
      v2f b = *(const v2f*)(kptr + stride2N * (size_t)(2 * j + khalf));
      acc = WMMA4(aq[j], b, acc);
    }
    float* tb = sTile[wave];
#pragma unroll
    for (int r = 0; r < 8; ++r) {
      int mm = r + halfsel;
      float v = acc[r];
      if (!mask[(size_t)(rowbase + mm) * Nn + (ncol + m)]) v = -1.0e9f;
      tb[mm * 17 + m] = v;
    }
    __syncthreads();
    if (lane < 16) {  // one lane per row: sorted-insert 16 candidates
      const int r = lane;
      float* tvr = sTV[wave][r];
      int* tir = sTI[wave][r];
      for (int j = 0; j < 16; ++j) {
        float v = tb[r * 17 + j];
        if (v > tvr[15]) {
          int idx = ncol + j;
          int pos = 15;
          while (pos > 0 && tvr[pos - 1] < v) {
            tvr[pos] = tvr[pos - 1];
            tir[pos] = tir[pos - 1];
            --pos;
          }
          tvr[pos] = v;
          tir[pos] = idx;
        }
      }
    }
    __syncthreads();
  }

  // merge 8 wave lists -> per-chunk top-16 per row
  if (tid < 16) {
    const int r = tid;
    float* mv = sMV[r];
    int* mi = sMI[r];
    for (int j = 0; j < 16; ++j) {
      mv[j] = -3.0e38f;
      mi[j] = 0;
    }
    for (int w = 0; w < 8; ++w) {
      const float* tvr = sTV[w][r];
      const int* tir = sTI[w][r];
      for (int j = 0; j < 16; ++j) {
        float v = tvr[j];
        if (v <= mv[15]) break;  // lists are sorted descending
        int idx = tir[j];
        int pos = 15;
        while (pos > 0 && (mv[pos - 1] < v || (mv[pos - 1] == v && mi[pos - 1] > idx))) {
          mv[pos] = mv[pos - 1];
          mi[pos] = mi[pos - 1];
          --pos;
        }
        mv[pos] = v;
        mi[pos] = idx;
      }
    }
    size_t base = ((size_t)(rowbase + r) * NCHUNK + blockIdx.y) * 16;
    for (int j = 0; j < 16; ++j) {
      candV[base + j] = mv[j];
      candI[base + j] = mi[j];
    }
  }
}

// ---------------------------------------------------------------------------
// Per-row tail: final top-k merge, gather, MLPs (WMMA), softmax, r_i.
// one block (256 thr) per query row.
// ---------------------------------------------------------------------------
__global__ __launch_bounds__(256) void tail_kernel(
    const float* __restrict__ qn, const float* __restrict__ candV,
    const int* __restrict__ candI, const float* __restrict__ bank_y,
    const float* __restrict__ Ws1, const float* __restrict__ bs1,
    const float* __restrict__ Ws2, const float* __restrict__ bs2,
    const float* __restrict__ Wc1, const float* __restrict__ bc1,
    const float* __restrict__ Wc2, const float* __restrict__ bc2,
    float* __restrict__ out_sim, float* __restrict__ out_idx,
    float* __restrict__ out_alpha, float* __restrict__ rbuf,
    float* __restrict__ hvv) {
  __shared__ float yb[16 * 64];
  __shared__ float hb[16 * 128];
  __shared__ float pj[16 * 128];
  __shared__ float tbuf[16 * 128];
  __shared__ float qrow[128];
  __shared__ float fv[16];
  __shared__ int fi[16];
  __shared__ float logitb[16];
  __shared__ float alphab[16];
  __shared__ float s_hv;

  const int row = blockIdx.x;
  const int tid = threadIdx.x;
  const int wave = tid >> 5;
  const int lane = tid & 31;
  const int m = lane & 15;
  const int kline = (lane >> 4) << 1;
  const int halfsel = (lane >> 4) << 3;

  if (tid == 0) {  // final merge of 32 sorted chunk lists
    for (int j = 0; j < 16; ++j) {
      fv[j] = -3.0e38f;
      fi[j] = 0;
    }
    for (int c = 0; c < NCHUNK; ++c) {
      size_t base = ((size_t)row * NCHUNK + c) * 16;
      for (int j = 0; j < 16; ++j) {
        float v = candV[base + j];
        if (v <= fv[15]) break;
        int idx = candI[base + j];
        int pos = 15;
        while (pos > 0 && (fv[pos - 1] < v || (fv[pos - 1] == v && fi[pos - 1] > idx))) {
          fv[pos] = fv[pos - 1];
          fi[pos] = fi[pos - 1];
          --pos;
        }
        fv[pos] = v;
        fi[pos] = idx;
      }
    }
    for (int j = 0; j < 16; ++j) {
      out_sim[(size_t)row * 16 + j] = fv[j];
      out_idx[(size_t)row * 16 + j] = (float)fi[j];
    }
    float hvf = (fv[0] > -5.0e8f) ? 1.0f : 0.0f;  // all-masked rows sit at -1e9
    s_hv = hvf;
    hvv[row] = hvf;
  }
  __syncthreads();

  for (int idx = tid; idx < 16 * 64; idx += 256)
    yb[idx] = bank_y[(size_t)fi[idx >> 6] * 64 + (idx & 63)];
  if (tid < 128) qrow[tid] = qn[(size_t)row * 128 + tid];
  __syncthreads();

  {  // h = relu(y @ Ws1^T + bs1)  : 16x128, K=64
    const int cbase = wave << 4;
    const float* wrow = Ws1 + (size_t)(cbase + m) * 64;
    const float* arow = yb + m * 64;
    v8f acc = {};
#pragma unroll
    for (int kk = 0; kk < 64; kk += 4) {
      v2f a = *(const v2f*)(arow + kk + kline);
      v2f b = *(const v2f*)(wrow + kk + kline);
      acc = WMMA4(a, b, acc);
    }
    float bv = bs1[cbase + m];
#pragma unroll
    for (int r = 0; r < 8; ++r)
      hb[(r + halfsel) * 128 + cbase + m] = fmaxf(acc[r] + bv, 0.0f);
  }
  __syncthreads();

  {  // proj_y = h @ Ws2^T + bs2 : 16x128, K=128
    const int cbase = wave << 4;
    const float* wrow = Ws2 + (size_t)(cbase + m) * 128;
    const float* arow = hb + m * 128;
    v8f acc = {};
#pragma unroll
    for (int kk = 0; kk < 128; kk += 4) {
      v2f a = *(const v2f*)(arow + kk + kline);
      v2f b = *(const v2f*)(wrow + kk + kline);
      acc = WMMA4(a, b, acc);
    }
    float bv = bs2[cbase + m];
#pragma unroll
    for (int r = 0; r < 8; ++r)
      pj[(r + halfsel) * 128 + cbase + m] = acc[r] + bv;
  }
  __syncthreads();

  {  // t = tanh([q_rep, proj_y] @ Wc1^T + bc1) : 16x128, K=256
    const int cbase = wave << 4;
    const float* wrow = Wc1 + (size_t)(cbase + m) * 256;
    const float* arow = pj + m * 128;
    v8f acc = {};
#pragma unroll
    for (int kk = 0; kk < 128; kk += 4) {  // q half: identical rows (broadcast)
      v2f a = *(const v2f*)(qrow + kk + kline);
      v2f b = *(const v2f*)(wrow + kk + kline);
      acc = WMMA4(a, b, acc);
    }
#pragma unroll
    for (int kk = 0; kk < 128; kk += 4) {  // proj_y half
      v2f a = *(const v2f*)(arow + kk + kline);
      v2f b = *(const v2f*)(wrow + 128 + kk + kline);
      acc = WMMA4(a, b, acc);
    }
    float bv = bc1[cbase + m];
#pragma unroll
    for (int r = 0; r < 8; ++r)
      tbuf[(r + halfsel) * 128 + cbase + m] = tanhf(acc[r] + bv);
  }
  __syncthreads();

  if (tid < 16) {
    float lg = bc2[0];
    for (int c = 0; c < 128; ++c) lg += tbuf[tid * 128 + c] * Wc2[c];
    logitb[tid] = lg;
  }
  __syncthreads();
  if (tid == 0) {
    if (s_hv != 0.0f) {
      float mx = logitb[0];
      for (int j = 1; j < 16; ++j) mx = fmaxf(mx, logitb[j]);
      float sum = 0.f;
      for (int j = 0; j < 16; ++j) {
        float e = expf(logitb[j] - mx);
        alphab[j] = e;
        sum += e;
      }
      float inv = 1.0f / sum;
      for (int j = 0; j < 16; ++j) alphab[j] *= inv;
    } else {
      for (int j = 0; j < 16; ++j) alphab[j] = 0.0f;
    }
    for (int j = 0; j < 16; ++j) out_alpha[(size_t)row * 16 + j] = alphab[j];
  }
  __syncthreads();
  if (tid < 128) {
    float r = 0.f;
    for (int j = 0; j < 16; ++j) r += alphab[j] * pj[j * 128 + tid];
    rbuf[(size_t)row * 128 + tid] = r;
  }
}

// ---------------------------------------------------------------------------
extern "C" void kernel_launch(void* const* d_in, const int* in_sizes, int n_in,
                              void* d_out, int out_size, void* d_ws,
                              size_t ws_size, hipStream_t stream) {
  (void)in_sizes; (void)n_in; (void)out_size; (void)ws_size;

  const float* z_i = (const float*)d_in[0];
  const float* g_i = (const float*)d_in[1];
  const float* bank_z = (const float*)d_in[2];
  const float* bank_g = (const float*)d_in[3];
  const float* bank_y = (const float*)d_in[4];
  const unsigned char* valid_mask = (const unsigned char*)d_in[5];
  const float* Wq = (const float*)d_in[6];
  const float* bq = (const float*)d_in[7];
  const float* Wk = (const float*)d_in[8];
  const float* bk = (const float*)d_in[9];
  const float* Ws1 = (const float*)d_in[10];
  const float* bs1 = (const float*)d_in[11];
  const float* Ws2 = (const float*)d_in[12];
  const float* bs2 = (const float*)d_in[13];
  const float* Wc1 = (const float*)d_in[14];
  const float* bc1 = (const float*)d_in[15];
  const float* Wc2 = (const float*)d_in[16];
  const float* bc2 = (const float*)d_in[17];
  const float* Wa = (const float*)d_in[18];
  const float* ba = (const float*)d_in[19];

  float* out = (float*)d_out;
  float* out_z = out;                         // [B,PD]
  float* out_sim = out_z + (size_t)Bq * PD;   // [B,16]
  float* out_idx = out_sim + (size_t)Bq * 16; // [B,16]
  float* out_alpha = out_idx + (size_t)Bq * 16;

  float* qn = (float*)d_ws;                       // [B,128]
  float* kP = qn + (size_t)Bq * RD;               // paired-transposed k: 128*N
  float* candV = kP + (size_t)RD * Nn;            // [B,NCHUNK,16]
  float* rbuf = candV + (size_t)Bq * NCHUNK * 16; // [B,128]
  float* hvv = rbuf + (size_t)Bq * RD;            // [B]
  int* candI = (int*)(hvv + Bq);                  // [B,NCHUNK,16]

  const size_t smemProj = (size_t)(16 * (PD + TD) + 16 * 128 + 16) * sizeof(float);
  const size_t smemAug = (size_t)(16 * (PD + RD)) * sizeof(float);

  // 1) q = l2norm(concat(z,g) @ Wq^T + bq)  -> row-major
  fused_linear_kernel<0><<<Bq / 16, 256, smemProj, stream>>>(
      z_i, PD, g_i, TD, Wq, bq, RD, Bq, qn, nullptr, nullptr, nullptr);
  // 2) k = l2norm(concat(bank_z,bank_g) @ Wk^T + bk) -> paired-transposed
  fused_linear_kernel<0><<<Nn / 16, 256, smemProj, stream>>>(
      bank_z, PD, bank_g, TD, Wk, bk, RD, Nn, nullptr, kP, nullptr, nullptr);
  // 3) fused sim GEMM + masked per-chunk top-16
  sim_topk_kernel<<<dim3(Bq / 16, NCHUNK), 256, 0, stream>>>(qn, kP, valid_mask,
                                                             candV, candI);
  // 4) final top-k, gather, MLPs, softmax, r_i
  tail_kernel<<<Bq, 256, 0, stream>>>(qn, candV, candI, bank_y, Ws1, bs1, Ws2,
                                      bs2, Wc1, bc1, Wc2, bc2, out_sim, out_idx,
                                      out_alpha, rbuf, hvv);
  // 5) z_tilde = where(hv, relu(concat(z,r) @ Wa^T + ba), z_i)
  fused_linear_kernel<1><<<Bq / 16, 256, smemAug, stream>>>(
      z_i, PD, rbuf, RD, Wa, ba, PD, Bq, out_z, nullptr, hvv, z_i);
}